// ModelGCN_77884936945814
// MI455X (gfx1250) — compile-verified
//
#include <hip/hip_runtime.h>

#define NUM_ROIS 148
#define BSZ 16
#define HID 64
#define IN_CH 128
#define OUT_CH 2
#define N1 131072
#define E1 2097152
#define N2 2368
#define E2 37888
#define NSEG (BSZ * NUM_ROIS)   /* 2368 */
#define EMB (NSEG * HID)        /* 151552 */
#define MLP_H 1000
#define LRELU 0.01f
#define BN_EPS 1e-5f

typedef __attribute__((ext_vector_type(2))) float v2f;
typedef __attribute__((ext_vector_type(8))) float v8f;

__global__ void fill_one(float* __restrict__ p, int n) {
  int i = blockIdx.x * blockDim.x + threadIdx.x;
  if (i < n) p[i] = 1.0f;
}

__global__ void deg_accum(const int* __restrict__ dst, const float* __restrict__ ew,
                          float* __restrict__ deg, int e_cnt) {
  int e = blockIdx.x * blockDim.x + threadIdx.x;
  if (e < e_cnt) atomicAdd(&deg[dst[e]], ew[e]);
}

__global__ void rsqrt_k(const float* __restrict__ deg, float* __restrict__ dinv, int n) {
  int i = blockIdx.x * blockDim.x + threadIdx.x;
  if (i < n) dinv[i] = rsqrtf(deg[i]);
}

// H[n x 64] = X[n x K] @ W[K x 64] via fp32 WMMA 16x16x4.
// One wave computes a full 16x64 tile (4 accumulators): A loaded once per K-step,
// 4 back-to-back WMMAs per A load (4x less A traffic, 4x WMMA density per wave).
// A layout: lane L holds row M=L%16, K-pair {0,1} (lanes 0-15) or {2,3} (16-31).
// B layout: lane L holds col N=L%16 of each sub-tile, same K-pair split.
// C/D layout: VGPR v -> row M = v + 8*(L/16), col N = L%16.
__global__ void gemm_wmma(const float* __restrict__ X, const float* __restrict__ W,
                          float* __restrict__ H, int K, int tiles_m) {
  int wave = (blockIdx.x * blockDim.x + threadIdx.x) >> 5;
  if (wave >= tiles_m) return;                 // wave-uniform: EXEC all-ones inside
  int lane = threadIdx.x & 31;
  int r = lane & 15;
  int half = lane >> 4;                        // 0 or 1 -> K offset 0 or 2
  const float* xrow = X + (wave * 16 + r) * K;
  v8f acc0 = {}, acc1 = {}, acc2 = {}, acc3 = {};
  for (int k = 0; k < K; k += 4) {
    v2f a = *(const v2f*)(xrow + k + 2 * half);
    const float* w0 = W + (k + 2 * half) * HID + r;
    const float* w1 = w0 + HID;
    v2f b0, b1, b2, b3;
    b0.x = w0[0];  b0.y = w1[0];
    b1.x = w0[16]; b1.y = w1[16];
    b2.x = w0[32]; b2.y = w1[32];
    b3.x = w0[48]; b3.y = w1[48];
    acc0 = __builtin_amdgcn_wmma_f32_16x16x4_f32(false, a, false, b0, (short)0, acc0,
                                                 false, false);
    acc1 = __builtin_amdgcn_wmma_f32_16x16x4_f32(false, a, false, b1, (short)0, acc1,
                                                 false, false);
    acc2 = __builtin_amdgcn_wmma_f32_16x16x4_f32(false, a, false, b2, (short)0, acc2,
                                                 false, false);
    acc3 = __builtin_amdgcn_wmma_f32_16x16x4_f32(false, a, false, b3, (short)0, acc3,
                                                 false, false);
  }
  float* hp = H + (wave * 16 + half * 8) * HID + r;
#pragma unroll
  for (int v = 0; v < 8; ++v) {
    hp[v * HID + 0]  = acc0[v];
    hp[v * HID + 16] = acc1[v];
    hp[v * HID + 32] = acc2[v];
    hp[v * HID + 48] = acc3[v];
  }
}

// out[dst] += dinv[src]*ew*dinv[dst] * h[src]; one thread per (edge, 4-channel chunk)
__global__ void edge_agg(const int* __restrict__ src, const int* __restrict__ dst,
                         const float* __restrict__ ew, const float* __restrict__ dinv,
                         const float* __restrict__ h, float* __restrict__ out, int e_cnt) {
  int idx = blockIdx.x * blockDim.x + threadIdx.x;
  int e = idx >> 4;
  if (e >= e_cnt) return;
  int c0 = (idx & 15) * 4;
  int s = src[e], d = dst[e];
  float norm = dinv[s] * ew[e] * dinv[d];
  float4 hv = *(const float4*)(h + s * HID + c0);
  float* od = out + d * HID + c0;
  atomicAdd(od + 0, norm * hv.x);
  atomicAdd(od + 1, norm * hv.y);
  atomicAdd(od + 2, norm * hv.z);
  atomicAdd(od + 3, norm * hv.w);
}

// out = relu(out + h * dinv^2 + bias)   (dinv^2 == 1/deg self-loop term)
__global__ void finish_relu(const float* __restrict__ h, const float* __restrict__ dinv,
                            const float* __restrict__ bias, float* __restrict__ out, int n64) {
  int idx = blockIdx.x * blockDim.x + threadIdx.x;
  if (idx >= n64) return;
  int i = idx >> 6, c = idx & 63;
  float di = dinv[i];
  float v = out[idx] + h[idx] * di * di + bias[c];
  out[idx] = fmaxf(v, 0.0f);
}

__global__ void pool_accum(const float* __restrict__ x, const int* __restrict__ roi,
                           const int* __restrict__ batch, float* __restrict__ sums,
                           float* __restrict__ cnt, int n) {
  int idx = blockIdx.x * blockDim.x + threadIdx.x;
  int i = idx >> 4;
  if (i >= n) return;
  int c0 = (idx & 15) * 4;
  int seg = batch[i] * NUM_ROIS + roi[i];
  float4 v = *(const float4*)(x + i * HID + c0);
  float* s = sums + seg * HID + c0;
  atomicAdd(s + 0, v.x);
  atomicAdd(s + 1, v.y);
  atomicAdd(s + 2, v.z);
  atomicAdd(s + 3, v.w);
  if (c0 == 0) atomicAdd(&cnt[seg], 1.0f);
}

__global__ void pool_final(const float* __restrict__ sums, const float* __restrict__ cnt,
                           float* __restrict__ emb) {
  int idx = blockIdx.x * blockDim.x + threadIdx.x;
  if (idx >= EMB) return;
  emb[idx] = sums[idx] / fmaxf(cnt[idx >> 6], 1.0f);
}

__global__ void add_emb(const float* __restrict__ a, const float* __restrict__ b,
                        float* __restrict__ s) {
  int idx = blockIdx.x * blockDim.x + threadIdx.x;
  if (idx < EMB) s[idx] = a[idx] + b[idx];
}

// h = LeakyReLU(BN(s @ Wm1 + bm1));  s: 16 x 9472, Wm1: 9472 x 1000
__global__ void mlp1(const float* __restrict__ s, const float* __restrict__ Wm1,
                     const float* __restrict__ bm1, const float* __restrict__ gamma,
                     const float* __restrict__ beta, const float* __restrict__ mean,
                     const float* __restrict__ var, float* __restrict__ hbuf) {
  int idx = blockIdx.x * blockDim.x + threadIdx.x;
  if (idx >= BSZ * MLP_H) return;
  int b = idx / MLP_H, j = idx % MLP_H;
  const float* srow = s + b * (NUM_ROIS * HID);
  float acc = 0.0f;
#pragma unroll 4
  for (int k = 0; k < NUM_ROIS * HID; ++k) acc = fmaf(srow[k], Wm1[k * MLP_H + j], acc);
  acc += bm1[j];
  acc = gamma[j] * (acc - mean[j]) * rsqrtf(var[j] + BN_EPS) + beta[j];
  hbuf[idx] = acc > 0.0f ? acc : LRELU * acc;
}

__global__ void mlp2(const float* __restrict__ hbuf, const float* __restrict__ Wm2,
                     const float* __restrict__ bm2, float* __restrict__ out) {
  int idx = blockIdx.x * blockDim.x + threadIdx.x;
  if (idx >= BSZ * OUT_CH) return;
  int b = idx / OUT_CH, j = idx % OUT_CH;
  float acc = 0.0f;
#pragma unroll 4
  for (int k = 0; k < MLP_H; ++k) acc = fmaf(hbuf[b * MLP_H + k], Wm2[k * OUT_CH + j], acc);
  out[idx] = acc + bm2[j];
}

static void run_branch(const float* x, int n, int e_cnt, const int* src, const int* dst,
                       const float* ew, const float* Wa, const float* ba,
                       const float* Wb, const float* bb, float* deg, float* dinv,
                       float* bufA, float* bufB, hipStream_t stream) {
  fill_one<<<(n + 255) / 256, 256, 0, stream>>>(deg, n);
  deg_accum<<<(e_cnt + 255) / 256, 256, 0, stream>>>(dst, ew, deg, e_cnt);
  rsqrt_k<<<(n + 255) / 256, 256, 0, stream>>>(deg, dinv, n);
  int tiles_m = n / 16;
  int gblocks = (tiles_m + 7) / 8;      // 8 waves (256 threads) per block, 1 tile/wave
  int eth = e_cnt * 16;
  int nh = n * HID;
  // layer 1: h = x @ Wa (WMMA), aggregate, self-loop + bias + relu
  gemm_wmma<<<gblocks, 256, 0, stream>>>(x, Wa, bufA, IN_CH, tiles_m);
  hipMemsetAsync(bufB, 0, (size_t)nh * sizeof(float), stream);
  edge_agg<<<(eth + 255) / 256, 256, 0, stream>>>(src, dst, ew, dinv, bufA, bufB, e_cnt);
  finish_relu<<<(nh + 255) / 256, 256, 0, stream>>>(bufA, dinv, ba, bufB, nh);
  // layer 2: h = relu1 @ Wb (WMMA), aggregate, self-loop + bias + relu
  gemm_wmma<<<gblocks, 256, 0, stream>>>(bufB, Wb, bufA, HID, tiles_m);
  hipMemsetAsync(bufB, 0, (size_t)nh * sizeof(float), stream);
  edge_agg<<<(eth + 255) / 256, 256, 0, stream>>>(src, dst, ew, dinv, bufA, bufB, e_cnt);
  finish_relu<<<(nh + 255) / 256, 256, 0, stream>>>(bufA, dinv, bb, bufB, nh);
  // result (n x 64) now in bufB
}

extern "C" void kernel_launch(void* const* d_in, const int* in_sizes, int n_in,
                              void* d_out, int out_size, void* d_ws, size_t ws_size,
                              hipStream_t stream) {
  (void)in_sizes; (void)n_in; (void)out_size; (void)ws_size;
  const float* x1   = (const float*)d_in[0];
  const int* nlab   = (const int*)d_in[1];
  const int* ei1    = (const int*)d_in[2];
  const float* ew1  = (const float*)d_in[3];
  const int* batch1 = (const int*)d_in[4];
  const float* x2   = (const float*)d_in[5];
  const int* rlab   = (const int*)d_in[6];
  const int* ei2    = (const int*)d_in[7];
  const float* ew2  = (const float*)d_in[8];
  const int* batch2 = (const int*)d_in[9];
  const float* W1a = (const float*)d_in[10]; const float* b1a = (const float*)d_in[11];
  const float* W1b = (const float*)d_in[12]; const float* b1b = (const float*)d_in[13];
  const float* W2a = (const float*)d_in[14]; const float* b2a = (const float*)d_in[15];
  const float* W2b = (const float*)d_in[16]; const float* b2b = (const float*)d_in[17];
  const float* Wm1 = (const float*)d_in[18]; const float* bm1 = (const float*)d_in[19];
  const float* gamma = (const float*)d_in[20]; const float* beta = (const float*)d_in[21];
  const float* bn_mean = (const float*)d_in[22]; const float* bn_var = (const float*)d_in[23];
  const float* Wm2 = (const float*)d_in[24]; const float* bm2 = (const float*)d_in[25];

  // workspace carve-up (floats)
  float* ws = (float*)d_ws;
  float* bufA = ws;  ws += (size_t)N1 * HID;   // 32 MB
  float* bufB = ws;  ws += (size_t)N1 * HID;   // 32 MB
  float* deg1 = ws;  ws += N1;
  float* dinv1 = ws; ws += N1;
  float* cA = ws;    ws += (size_t)N2 * HID;
  float* cB = ws;    ws += (size_t)N2 * HID;
  float* deg2 = ws;  ws += N2;
  float* dinv2 = ws; ws += N2;
  float* sums = ws;  ws += EMB;
  float* cnt = ws;   ws += NSEG;
  float* hbuf = ws;  ws += BSZ * MLP_H;

  // output carve-up: out(32) | embedding(151552) | embedding_roi(151552) | sum(151552)
  float* out_logits = (float*)d_out;
  float* emb1 = out_logits + BSZ * OUT_CH;
  float* emb2 = emb1 + EMB;
  float* embsum = emb2 + EMB;

  // ---- ROI branch (x2) ----
  run_branch(x2, N2, E2, ei2, ei2 + E2, ew2, W2a, b2a, W2b, b2b,
             deg2, dinv2, cA, cB, stream);
  hipMemsetAsync(sums, 0, (size_t)EMB * sizeof(float), stream);
  hipMemsetAsync(cnt, 0, (size_t)NSEG * sizeof(float), stream);
  pool_accum<<<(N2 * 16 + 255) / 256, 256, 0, stream>>>(cB, rlab, batch2, sums, cnt, N2);
  pool_final<<<(EMB + 255) / 256, 256, 0, stream>>>(sums, cnt, emb2);

  // ---- fine-grained branch (x1) ----
  run_branch(x1, N1, E1, ei1, ei1 + E1, ew1, W1a, b1a, W1b, b1b,
             deg1, dinv1, bufA, bufB, stream);
  hipMemsetAsync(sums, 0, (size_t)EMB * sizeof(float), stream);
  hipMemsetAsync(cnt, 0, (size_t)NSEG * sizeof(float), stream);
  pool_accum<<<(N1 * 16 + 255) / 256, 256, 0, stream>>>(bufB, nlab, batch1, sums, cnt, N1);
  pool_final<<<(EMB + 255) / 256, 256, 0, stream>>>(sums, cnt, emb1);

  // ---- classifier ----
  add_emb<<<(EMB + 255) / 256, 256, 0, stream>>>(emb1, emb2, embsum);
  mlp1<<<(BSZ * MLP_H + 255) / 256, 256, 0, stream>>>(embsum, Wm1, bm1, gamma, beta,
                                                      bn_mean, bn_var, hbuf);
  mlp2<<<1, 32, 0, stream>>>(hbuf, Wm2, bm2, out_logits);
}